// DNCDynamics_62964220559493
// MI455X (gfx1250) — compile-verified
//
#include <hip/hip_runtime.h>
#include <hip/hip_bf16.h>

// ---- problem constants (from reference) ----
#define Bv   64
#define Lv   64
#define INv  1024
#define ACTv 128
#define Hv   512
#define Nv   128
#define Wv   64
#define Rv   4
#define OUTv 1024
#define IFv  471
#define CINv 1408           // IN + ACT + R*W

#define MB     16           // batches per block
#define NT     512          // threads per block (16 wave32)
#define NWAVE  16

// ---- per-batch state layout in d_ws (floats) ----
#define ST_M    0
#define ST_LINK (ST_M + Nv*Wv)        // 8192
#define ST_U    (ST_LINK + Nv*Nv)     // 24576
#define ST_PREC (ST_U + Nv)
#define ST_WW   (ST_PREC + Nv)
#define ST_WR   (ST_WW + Nv)
#define ST_RV   (ST_WR + Rv*Nv)
#define ST_SZ   (ST_RV + Rv*Wv)       // 25728 floats/batch

// ---- LDS layout (floats), padded strides to dodge bank conflicts ----
#define CINP 1412
#define HP   516
#define IFP  484
#define RVP  260
#define O_SA   0
#define O_SORT (O_SA   + MB*CINP)
#define O_CP   (O_SORT + MB*Nv)
#define O_H    (O_CP   + MB*Nv)
#define O_XI   (O_H    + MB*HP)
#define O_ER   (O_XI   + MB*IFP)
#define O_WV   (O_ER   + MB*Wv)
#define O_WW   (O_WV   + MB*Wv)
#define O_WRB  (O_WW   + MB*Nv)
#define O_RVB  (O_WRB  + MB*Rv*Nv)
#define LDS_FLOATS (O_RVB + MB*RVP)   // 59136 floats = 236544 B (< 320KB/WGP)

typedef __attribute__((ext_vector_type(2))) float v2f;
typedef __attribute__((ext_vector_type(8))) float v8f;

__device__ __forceinline__ v8f wmma4(v2f a, v2f b, v8f c) {
  // D = A(16x4) * B(4x16) + C(16x16), full fp32
  return __builtin_amdgcn_wmma_f32_16x16x4_f32(false, a, false, b, (short)0, c,
                                               false, false);
}

__device__ __forceinline__ float wsum32(float v) {
  #pragma unroll
  for (int o = 16; o > 0; o >>= 1) v += __shfl_xor(v, o, 32);
  return v;
}
__device__ __forceinline__ float wmax32(float v) {
  #pragma unroll
  for (int o = 16; o > 0; o >>= 1) v = fmaxf(v, __shfl_xor(v, o, 32));
  return v;
}
__device__ __forceinline__ float sigm(float x)   { return 1.f / (1.f + __expf(-x)); }
__device__ __forceinline__ float oneplus(float x){ return (x > 20.f) ? (1.f + x) : (1.f + log1pf(__expf(x))); }
// branch-free tanh: avoids libm's exec-mask save/restore in the hot loop
__device__ __forceinline__ float fast_tanh(float x) {
  float t = __expf(-2.f * fabsf(x));          // in (0,1], no overflow
  float r = (1.f - t) / (1.f + t);
  return copysignf(r, x);
}

__global__ void __launch_bounds__(NT, 1)
dnc_persistent_kernel(const float* __restrict__ embed,
                      const float* __restrict__ actions,
                      const float* __restrict__ Wc, const float* __restrict__ bc,
                      const float* __restrict__ Wi, const float* __restrict__ bi,
                      const float* __restrict__ Wy, const float* __restrict__ Wr,
                      const float* __restrict__ by,
                      float* __restrict__ out, float* __restrict__ ws)
{
  extern __shared__ float sm[];
  const int tid  = threadIdx.x;
  const int wave = tid >> 5;
  const int lane = tid & 31;
  const int bg   = blockIdx.x;                    // batch group (4 groups of 16)
  float* wsblk = ws + (size_t)bg * MB * ST_SZ;

  // zero-init recurrent state (deterministic per launch)
  for (int i = tid; i < MB * ST_SZ; i += NT) wsblk[i] = 0.f;
  __syncthreads();

  const int m16  = lane & 15;                     // WMMA row (A) / col (B,D)
  const int koff = (lane >> 4) << 1;              // 0 or 2 (K sub-offset)
  const int grow = (lane >> 4) << 3;              // D row base: 0 or 8

  for (int t = 0; t < Lv; ++t) {
    // ================= Phase A: stage ctrl_in = [x | rv] into LDS ==========
    for (int i = tid; i < MB * CINv; i += NT) {
      int row = i / CINv, c = i - row * CINv;
      int b = bg * MB + row;
      float v;
      if (c < INv)                 v = embed[((size_t)b * Lv + t) * INv + c];
      else if (c < INv + ACTv)     v = actions[((size_t)b * Lv + t) * ACTv + (c - INv)];
      else                         v = wsblk[row * ST_SZ + ST_RV + (c - INv - ACTv)];
      sm[O_SA + row * CINP + c] = v;
    }
    __syncthreads();

    // ================= Phase B: h = tanh(ctrl_in @ Wc + bc) ================
    #pragma unroll
    for (int tt = 0; tt < 2; ++tt) {
      int n0 = (wave + tt * NWAVE) * 16;
      int nc = n0 + m16;
      v8f acc = {};
      const float* Ab = sm + O_SA + m16 * CINP;
      for (int k = 0; k < CINv; k += 4) {
        v2f a, b;
        a.x = Ab[k + koff]; a.y = Ab[k + koff + 1];
        const float* bp = Wc + (size_t)(k + koff) * Hv + nc;
        b.x = bp[0]; b.y = bp[Hv];
        acc = wmma4(a, b, acc);
      }
      float bias = bc[nc];
      #pragma unroll
      for (int g = 0; g < 8; ++g)
        sm[O_H + (g + grow) * HP + nc] = fast_tanh(acc[g] + bias);
    }
    __syncthreads();

    // ================= Phase C: xi = h @ Wi + bi (471 cols) ================
    #pragma unroll
    for (int tt = 0; tt < 2; ++tt) {
      int tile = wave + tt * NWAVE;
      if (tile < 30) {
        int nc = tile * 16 + m16;
        // clamp: out-of-range columns compute garbage that is never stored
        int ncl = (nc < IFv) ? nc : (IFv - 1);
        v8f acc = {};
        const float* Ab = sm + O_H + m16 * HP;
        for (int k = 0; k < Hv; k += 4) {
          v2f a, b;
          a.x = Ab[k + koff]; a.y = Ab[k + koff + 1];
          const float* bp = Wi + (size_t)(k + koff) * IFv + ncl;
          b.x = bp[0]; b.y = bp[IFv];
          acc = wmma4(a, b, acc);
        }
        if (nc < IFv) {
          float bias = bi[nc];
          #pragma unroll
          for (int g = 0; g < 8; ++g)
            sm[O_XI + (g + grow) * IFP + nc] = acc[g] + bias;
        }
      }
    }
    __syncthreads();

    // ================= Phase D: DNC memory update (wave -> batch) ==========
    {
      float*       st  = wsblk + wave * ST_SZ;
      const float* xi  = sm + O_XI + wave * IFP;
      float*       uS  = sm + O_SORT + wave * Nv;
      float*       cpS = sm + O_CP   + wave * Nv;
      float*       erS = sm + O_ER   + wave * Wv;
      float*       wvS = sm + O_WV   + wave * Wv;
      float*       wwS = sm + O_WW   + wave * Nv;
      float*       wrB = sm + O_WRB  + wave * Rv * Nv;   // prev then new read wts
      float*       rvB = sm + O_RVB  + wave * RVP;

      // interface scalars (xi splits: rk 0 | rb 256 | wk 260 | wb 324 | er 325
      //                    | wv 389 | fg 453 | ga 457 | gw 458 | modes 459)
      float freeg[Rv], rbeta[Rv];
      #pragma unroll
      for (int r = 0; r < Rv; ++r) { freeg[r] = sigm(xi[453 + r]); rbeta[r] = oneplus(xi[256 + r]); }
      float wbeta = oneplus(xi[324]);
      float g_a = sigm(xi[457]);
      float g_w = sigm(xi[458]);

      #pragma unroll
      for (int q = 0; q < 2; ++q) {
        int w = lane + 32 * q;
        erS[w] = sigm(xi[325 + w]);
        wvS[w] = xi[389 + w];
      }

      // stage wr_prev into LDS (and regs for owned slots)
      float wrp[Rv][4];
      #pragma unroll
      for (int q = 0; q < 4; ++q) {
        int n = lane + 32 * q;
        #pragma unroll
        for (int r = 0; r < Rv; ++r) {
          float v = st[ST_WR + r * Nv + n];
          wrp[r][q] = v;
          wrB[r * Nv + n] = v;
        }
      }

      // ---- usage update: u = (u + ww - u*ww) * psi ----
      float uq[4], precq[4], wwq[4];
      #pragma unroll
      for (int q = 0; q < 4; ++q) {
        int n = lane + 32 * q;
        float psi = 1.f;
        #pragma unroll
        for (int r = 0; r < Rv; ++r) psi *= (1.f - freeg[r] * wrp[r][q]);
        float u0 = st[ST_U + n], w0 = st[ST_WW + n];
        uq[q] = (u0 + w0 - u0 * w0) * psi;
        st[ST_U + n] = uq[q];
        precq[q] = st[ST_PREC + n];
        uS[n] = uq[q];
      }
      __syncthreads();

      // ---- allocation via rank + exclusive cumprod over sorted usage ----
      int rank[4] = {0, 0, 0, 0};
      for (int mth = 0; mth < Nv; ++mth) {
        float um = uS[mth];
        #pragma unroll
        for (int q = 0; q < 4; ++q) {
          int n = lane + 32 * q;
          rank[q] += (um < uq[q]) || (um == uq[q] && mth < n);
        }
      }
      #pragma unroll
      for (int q = 0; q < 4; ++q) cpS[rank[q]] = uq[q];
      __syncthreads();
      if (lane == 0) {
        float run = 1.f;
        for (int j = 0; j < Nv; ++j) { float tv = cpS[j]; cpS[j] = run; run *= tv; }
      }
      __syncthreads();
      float alloc[4];
      #pragma unroll
      for (int q = 0; q < 4; ++q) alloc[q] = (1.f - uq[q]) * cpS[rank[q]];

      // ---- write content weights (cosine, old M) + write weighting ----
      float kn2 = 0.f;
      for (int w = 0; w < Wv; ++w) { float kv = xi[260 + w]; kn2 += kv * kv; }
      float knI = 1.f / (sqrtf(kn2) + 1e-6f);
      float simq[4];
      #pragma unroll
      for (int q = 0; q < 4; ++q) {
        int n = lane + 32 * q;
        const float* Mr = st + ST_M + n * Wv;
        float dot = 0.f, nm2 = 0.f;
        for (int w = 0; w < Wv; ++w) { float mv = Mr[w]; dot += mv * xi[260 + w]; nm2 += mv * mv; }
        simq[q] = wbeta * dot * knI / (sqrtf(nm2) + 1e-6f);
      }
      float mx = wmax32(fmaxf(fmaxf(simq[0], simq[1]), fmaxf(simq[2], simq[3])));
      float es[4], ssum = 0.f;
      #pragma unroll
      for (int q = 0; q < 4; ++q) { es[q] = __expf(simq[q] - mx); ssum += es[q]; }
      ssum = wsum32(ssum);
      float sinv = 1.f / ssum;
      float wwloc = 0.f;
      #pragma unroll
      for (int q = 0; q < 4; ++q) {
        float cw = es[q] * sinv;
        wwq[q] = g_w * (g_a * alloc[q] + (1.f - g_a) * cw);
        wwloc += wwq[q];
        int n = lane + 32 * q;
        wwS[n] = wwq[q];
        st[ST_WW + n] = wwq[q];
      }
      float wwtot = wsum32(wwloc);
      __syncthreads();

      // ---- memory write: M = M*(1 - ww*erase) + ww*write_vec ----
      #pragma unroll
      for (int q = 0; q < 4; ++q) {
        int n = lane + 32 * q;
        float wn = wwq[q];
        float* Mr = st + ST_M + n * Wv;
        for (int w = 0; w < Wv; ++w)
          Mr[w] = Mr[w] * (1.f - wn * erS[w]) + wn * wvS[w];
      }

      // ---- link update (column-partitioned) fused with bwd ----
      float bwd[Rv][4], fwd[Rv][4];
      #pragma unroll
      for (int r = 0; r < Rv; ++r)
        #pragma unroll
        for (int q = 0; q < 4; ++q) { bwd[r][q] = 0.f; fwd[r][q] = 0.f; }
      for (int i = 0; i < Nv; ++i) {
        float wwi = wwS[i];
        float wrpI[Rv];
        #pragma unroll
        for (int r = 0; r < Rv; ++r) wrpI[r] = wrB[r * Nv + i];
        #pragma unroll
        for (int q = 0; q < 4; ++q) {
          int j = lane + 32 * q;
          float lo = st[ST_LINK + i * Nv + j];
          float ln = (i == j) ? 0.f : ((1.f - wwi - wwq[q]) * lo + wwi * precq[q]);
          st[ST_LINK + i * Nv + j] = ln;
          #pragma unroll
          for (int r = 0; r < Rv; ++r) bwd[r][q] += ln * wrpI[r];
        }
      }
      // precedence: prec = (1 - sum(ww)) * prec + ww
      #pragma unroll
      for (int q = 0; q < 4; ++q) {
        int n = lane + 32 * q;
        st[ST_PREC + n] = (1.f - wwtot) * precq[q] + wwq[q];
      }
      __syncthreads();   // updated link visible across lanes

      // ---- fwd (row-partitioned on updated link) ----
      #pragma unroll
      for (int q = 0; q < 4; ++q) {
        int i = lane + 32 * q;
        const float* lrow = st + ST_LINK + i * Nv;
        for (int j = 0; j < Nv; ++j) {
          float ln = lrow[j];
          #pragma unroll
          for (int r = 0; r < Rv; ++r) fwd[r][q] += ln * wrB[r * Nv + j];
        }
      }

      // ---- read content weights (new M) + combine via modes ----
      #pragma unroll
      for (int r = 0; r < Rv; ++r) {
        float kn2r = 0.f;
        for (int w = 0; w < Wv; ++w) { float kv = xi[r * Wv + w]; kn2r += kv * kv; }
        float knIr = 1.f / (sqrtf(kn2r) + 1e-6f);
        float sq[4];
        #pragma unroll
        for (int q = 0; q < 4; ++q) {
          int n = lane + 32 * q;
          const float* Mr = st + ST_M + n * Wv;
          float dot = 0.f, nm2 = 0.f;
          for (int w = 0; w < Wv; ++w) { float mv = Mr[w]; dot += mv * xi[r * Wv + w]; nm2 += mv * mv; }
          sq[q] = rbeta[r] * dot * knIr / (sqrtf(nm2) + 1e-6f);
        }
        float mr_ = wmax32(fmaxf(fmaxf(sq[0], sq[1]), fmaxf(sq[2], sq[3])));
        float er_[4], ss = 0.f;
        #pragma unroll
        for (int q = 0; q < 4; ++q) { er_[q] = __expf(sq[q] - mr_); ss += er_[q]; }
        ss = wsum32(ss);
        float ssI = 1.f / ss;
        // modes softmax (3-way)
        float m0 = xi[459 + r * 3], m1 = xi[459 + r * 3 + 1], m2 = xi[459 + r * 3 + 2];
        float mm = fmaxf(m0, fmaxf(m1, m2));
        float e0 = __expf(m0 - mm), e1 = __expf(m1 - mm), e2 = __expf(m2 - mm);
        float inv = 1.f / (e0 + e1 + e2);
        float p0 = e0 * inv, p1 = e1 * inv, p2 = e2 * inv;
        #pragma unroll
        for (int q = 0; q < 4; ++q) {
          int n = lane + 32 * q;
          float cr = er_[q] * ssI;
          float wrn = p0 * bwd[r][q] + p1 * cr + p2 * fwd[r][q];
          st[ST_WR + r * Nv + n] = wrn;
          wrB[r * Nv + n] = wrn;               // overwrite prev (done with it)
        }
      }
      __syncthreads();

      // ---- read vectors: rv[r][w] = sum_n wr[r][n] * M[n][w] ----
      #pragma unroll
      for (int q = 0; q < 2; ++q) {
        int w = lane + 32 * q;
        float acc[Rv] = {0.f, 0.f, 0.f, 0.f};
        for (int n = 0; n < Nv; ++n) {
          float mv = st[ST_M + n * Wv + w];
          #pragma unroll
          for (int r = 0; r < Rv; ++r) acc[r] += wrB[r * Nv + n] * mv;
        }
        #pragma unroll
        for (int r = 0; r < Rv; ++r) {
          st[ST_RV + r * Wv + w] = acc[r];
          rvB[r * Wv + w] = acc[r];
        }
      }
    }
    __syncthreads();

    // ================= Phase E: out = h @ Wy + rv @ Wr + by ================
    #pragma unroll
    for (int tt = 0; tt < 4; ++tt) {
      int nc = (wave + tt * NWAVE) * 16 + m16;
      v8f acc = {};
      const float* Ah = sm + O_H + m16 * HP;
      for (int k = 0; k < Hv; k += 4) {
        v2f a, b;
        a.x = Ah[k + koff]; a.y = Ah[k + koff + 1];
        const float* bp = Wy + (size_t)(k + koff) * OUTv + nc;
        b.x = bp[0]; b.y = bp[OUTv];
        acc = wmma4(a, b, acc);
      }
      const float* Ar = sm + O_RVB + m16 * RVP;
      for (int k = 0; k < Rv * Wv; k += 4) {
        v2f a, b;
        a.x = Ar[k + koff]; a.y = Ar[k + koff + 1];
        const float* bp = Wr + (size_t)(k + koff) * OUTv + nc;
        b.x = bp[0]; b.y = bp[OUTv];
        acc = wmma4(a, b, acc);
      }
      float bias = by[nc];
      #pragma unroll
      for (int g = 0; g < 8; ++g) {
        int b = bg * MB + (g + grow);
        out[((size_t)b * Lv + t) * OUTv + nc] = acc[g] + bias;
      }
    }
    __syncthreads();
  }
}

extern "C" void kernel_launch(void* const* d_in, const int* in_sizes, int n_in,
                              void* d_out, int out_size, void* d_ws, size_t ws_size,
                              hipStream_t stream) {
  const float* embed   = (const float*)d_in[0];
  const float* actions = (const float*)d_in[1];
  const float* Wc      = (const float*)d_in[2];
  const float* bc      = (const float*)d_in[3];
  const float* Wi      = (const float*)d_in[4];
  const float* bi      = (const float*)d_in[5];
  const float* Wy      = (const float*)d_in[6];
  const float* Wr      = (const float*)d_in[7];
  const float* by      = (const float*)d_in[8];
  float* out = (float*)d_out;
  float* ws  = (float*)d_ws;   // needs B*ST_SZ floats (~6.6 MB), zeroed in-kernel

  dim3 grid(Bv / MB);          // 4 persistent workgroups, one per 16-batch group
  dim3 block(NT);              // 512 threads = 16 wave32
  size_t lds_bytes = (size_t)LDS_FLOATS * sizeof(float);   // 236544 B
  dnc_persistent_kernel<<<grid, block, lds_bytes, stream>>>(
      embed, actions, Wc, bc, Wi, bi, Wy, Wr, by, out, ws);
}